// SOCA_15796889714760
// MI455X (gfx1250) — compile-verified
//
#include <hip/hip_runtime.h>
#include <hip/hip_bf16.h>
#include <math.h>

typedef __attribute__((ext_vector_type(2))) float v2f;
typedef __attribute__((ext_vector_type(8))) float v8f;

#define LP 66            // LDS row pitch (floats): conflict-free A reads, 8B-aligned pairs
#define CC 64            // channels
#define MM 2304          // 48*48
#define NB 256           // batch

// D = A(16x4) x B(4x16) + C fp32 WMMA over a 64x64 LDS matmul.
// A, B are 64x64 matrices stored with row pitch LP in LDS.
// Wave tile: rows [c0,c0+16), cols [d0,d0+16). lane: l = lane&15, half = lane>>4.
__device__ __forceinline__ v8f mm64(const float* __restrict__ A, const float* __restrict__ B,
                                    int c0, int d0, int l, int half) {
  v8f acc = {};
  const float* arow = A + (c0 + l) * LP + 2 * half;        // A[M=c0+l][K=2*half + {0,1}]
  const float* bcol = B + 2 * half * LP + (d0 + l);        // B[K=2*half + {0,1}][N=d0+l]
#pragma unroll
  for (int k0 = 0; k0 < 64; k0 += 4) {
    v2f a = *(const v2f*)(arow + k0);
    v2f b;
    b.x = bcol[k0 * LP];
    b.y = bcol[k0 * LP + LP];
    acc = __builtin_amdgcn_wmma_f32_16x16x4_f32(false, a, false, b, (short)0, acc, false, false);
  }
  return acc;
}

// Store a 16x16 f32 C/D fragment: VGPR r -> M = r + 8*half, N = l (per ISA layout).
__device__ __forceinline__ void store_tile(float* __restrict__ D, const v8f& acc,
                                           int c0, int d0, int l, int half) {
  int N = d0 + l;
#pragma unroll
  for (int r = 0; r < 8; ++r) D[(c0 + 8 * half + r) * LP + N] = acc[r];
}

__global__ void __launch_bounds__(512)
soca_batch_kernel(const float* __restrict__ x,
                  const float* __restrict__ w1, const float* __restrict__ b1,
                  const float* __restrict__ w2, const float* __restrict__ b2,
                  float* __restrict__ att) {
  __shared__ float sY[CC * LP];
  __shared__ float sZ[CC * LP];
  __shared__ float sZY[CC * LP];
  __shared__ float sPart[2][CC];
  __shared__ float sMu[CC];
  __shared__ float sVec[CC];
  __shared__ float sH[CC];
  __shared__ float sScal[2];   // [0]=1/trace, [1]=sqrt(trace)

  const int b    = blockIdx.x;
  const int tid  = threadIdx.x;
  const int wv   = tid >> 5;
  const int lane = tid & 31;
  const int half = lane >> 4;
  const int l    = lane & 15;
  const int c0   = (wv >> 2) << 4;   // NS-phase tile (all 16 waves)
  const int d0   = (wv & 3) << 4;

  // Lower-triangle tile decode for the Gram phase (waves 0..9): packed nibble LUTs.
  // wv:  0  1  2  3  4  5  6  7  8  9
  // gi:  0  1  1  2  2  2  3  3  3  3
  // gj:  0  0  1  0  1  2  0  1  2  3
  const int gi = (int)((0x3333222110ULL >> (4 * wv)) & 0xF) << 4;
  const int gj = (int)((0x3210210100ULL >> (4 * wv)) & 0xF) << 4;

  // ---- Phase 1: symmetric Gram matrix G = X X^T via fp32 WMMA (10 unique tiles) ----
  if (wv < 10) {
    const float* xb   = x + (size_t)b * CC * MM;
    const float* aptr = xb + (size_t)(gi + l) * MM + 2 * half;
    const float* bptr = xb + (size_t)(gj + l) * MM + 2 * half;
    v8f g = {};
    float rs = 0.f;  // row-sum partials (gj==0 waves cover every row as A-rows)
#pragma unroll 4
    for (int k0 = 0; k0 < MM; k0 += 4) {
      v2f a  = *(const v2f*)(aptr + k0);
      v2f bb = *(const v2f*)(bptr + k0);
      rs += a.x + a.y;
      g = __builtin_amdgcn_wmma_f32_16x16x4_f32(false, a, false, bb, (short)0, g, false, false);
    }
    if (gj == 0) sPart[half][gi + l] = rs;
    __syncthreads();

    if (tid < CC) sMu[tid] = (sPart[0][tid] + sPart[1][tid]) * (1.0f / MM);
    __syncthreads();

    // cov = G/m - mu mu^T -> sZ (tile + mirrored transpose for off-diagonal tiles)
    {
      const float invm = 1.0f / MM;
      int N = gj + l;
      float muN = sMu[N];
#pragma unroll
      for (int r = 0; r < 8; ++r) {
        int Mrow = gi + 8 * half + r;
        float v = g[r] * invm - sMu[Mrow] * muN;
        sZ[Mrow * LP + N] = v;
        if (gi != gj) sZ[N * LP + Mrow] = v;   // symmetry mirror
      }
    }
  } else {
    __syncthreads();   // match the two barriers taken by active waves
    __syncthreads();
  }
  __syncthreads();

  if (tid == 0) {
    float tr = 0.f;
    for (int i = 0; i < CC; ++i) tr += sZ[i * LP + i];
    sScal[0] = 1.0f / tr;
    sScal[1] = sqrtf(tr);
  }
  __syncthreads();

  // A = cov/tr (in sZ);  ZY0 = 0.5*(3I - A) (in sZY)
  {
    const float invTr = sScal[0];
    for (int idx = tid; idx < CC * CC; idx += 512) {
      int r = idx >> 6, c = idx & 63;
      float aV = sZ[r * LP + c] * invTr;
      sZ[r * LP + c]  = aV;
      sZY[r * LP + c] = ((r == c) ? 1.5f : 0.0f) - 0.5f * aV;
    }
  }
  __syncthreads();

  // Y1 = A @ ZY0 ; Z1 = ZY0
  {
    v8f y1 = mm64(sZ, sZY, c0, d0, l, half);
    __syncthreads();
    store_tile(sY, y1, c0, d0, l, half);
    for (int idx = tid; idx < CC * CC; idx += 512) {
      int r = idx >> 6, c = idx & 63;
      sZ[r * LP + c] = sZY[r * LP + c];
    }
    __syncthreads();
  }

  // 3 Newton-Schulz iterations
  for (int it = 0; it < 3; ++it) {
    v8f t = mm64(sZ, sY, c0, d0, l, half);                 // Z @ Y
    __syncthreads();
    {
      int N = d0 + l;
#pragma unroll
      for (int r = 0; r < 8; ++r) {
        int Mrow = c0 + 8 * half + r;
        sZY[Mrow * LP + N] = ((Mrow == N) ? 1.5f : 0.0f) - 0.5f * t[r];
      }
    }
    __syncthreads();
    v8f yn = mm64(sY, sZY, c0, d0, l, half);               // Y @ ZY
    v8f zn = mm64(sZY, sZ, c0, d0, l, half);               // ZY @ Z
    __syncthreads();
    store_tile(sY, yn, c0, d0, l, half);
    store_tile(sZ, zn, c0, d0, l, half);
    __syncthreads();
  }

  // Final: U = 3I - Z@Y ; Yf = Y @ U ; y = 0.5*sqrt(tr)*Yf ; vec = column-mean(y)
  {
    v8f t = mm64(sZ, sY, c0, d0, l, half);
    __syncthreads();
    int N = d0 + l;
#pragma unroll
    for (int r = 0; r < 8; ++r) {
      int Mrow = c0 + 8 * half + r;
      sZY[Mrow * LP + N] = ((Mrow == N) ? 3.0f : 0.0f) - t[r];
    }
    __syncthreads();
    v8f f = mm64(sY, sZY, c0, d0, l, half);
    __syncthreads();
    store_tile(sZY, f, c0, d0, l, half);
    __syncthreads();
  }

  const float sqrtTr = sScal[1];
  if (tid < CC) {
    float s = 0.f;
    for (int c = 0; c < CC; ++c) s += sZY[c * LP + tid];
    sVec[tid] = s * (0.5f * sqrtTr * (1.0f / CC));
  }
  __syncthreads();

  // conv1 (center tap only, since spatial is 1x1): h = relu(W1c @ vec + b1)
  if (tid < CC) {
    float acc = b1[tid];
    for (int i = 0; i < CC; ++i) acc += w1[(tid * CC + i) * 9 + 4] * sVec[i];
    sH[tid] = fmaxf(acc, 0.0f);
  }
  __syncthreads();

  // conv2: att = sigmoid(W2c @ h + b2)
  if (tid < CC) {
    float acc = b2[tid];
    for (int i = 0; i < CC; ++i) acc += w2[(tid * CC + i) * 9 + 4] * sH[i];
    att[b * CC + tid] = 1.0f / (1.0f + expf(-acc));
  }
}

// out[b,c,:,:] = att[b,c] * x[b,c,:,:].
// One block per (b,c) plane: att index is block-uniform (scalar load), 576 float4s per plane.
__global__ void __launch_bounds__(576)
soca_scale_kernel(const float* __restrict__ x, const float* __restrict__ att,
                  float* __restrict__ out) {
  const size_t bc = blockIdx.x;
  const float a = att[bc];                       // uniform -> scalar load
  const size_t i = bc * (MM / 4) + threadIdx.x;  // float4 index
  float4 xv = ((const float4*)x)[i];
  float4 o;
  o.x = xv.x * a; o.y = xv.y * a; o.z = xv.z * a; o.w = xv.w * a;
  ((float4*)out)[i] = o;
}

extern "C" void kernel_launch(void* const* d_in, const int* in_sizes, int n_in,
                              void* d_out, int out_size, void* d_ws, size_t ws_size,
                              hipStream_t stream) {
  (void)in_sizes; (void)n_in; (void)out_size; (void)ws_size;
  const float* x  = (const float*)d_in[0];
  const float* w1 = (const float*)d_in[1];
  const float* b1 = (const float*)d_in[2];
  const float* w2 = (const float*)d_in[3];
  const float* b2 = (const float*)d_in[4];
  float* out = (float*)d_out;
  float* att = (float*)d_ws;   // NB*CC floats

  soca_batch_kernel<<<NB, 512, 0, stream>>>(x, w1, b1, w2, b2, att);
  soca_scale_kernel<<<NB * CC, MM / 4, 0, stream>>>(x, att, out);
}